// HiddenStateMLPPooling_34772055228496
// MI455X (gfx1250) — compile-verified
//
#include <hip/hip_runtime.h>
#include <math.h>

typedef __attribute__((ext_vector_type(2))) float v2f;
typedef __attribute__((ext_vector_type(8))) float v8f;

#define FILLV   (-100.0f)
#define NEG_INF (-3.402823466e+38f)

// ---------------------------------------------------------------------------
// Kernel 1: pairwise spatial/velocity branches, collapsed from O(N^2) to O(N).
// pooled cols [0,32)  = relu(max_j p_s(j) - p_s(i) + b_s)   (or FILL if row i NaN)
// pooled cols [96,128)= relu(max_j p_v(j) - p_v(i) + b_v)   (or FILL)
// ---------------------------------------------------------------------------
__global__ void prep_pairwise(const float* __restrict__ obs1,
                              const float* __restrict__ obs2,
                              const float* __restrict__ W_s, const float* __restrict__ b_s,
                              const float* __restrict__ W_v, const float* __restrict__ b_v,
                              float* __restrict__ pooled)
{
    const int b = blockIdx.x;          // batch
    const int t = threadIdx.x;         // 0..255
    const int N = 512;

    __shared__ float sPart[8][32];
    __shared__ float vPart[8][32];
    __shared__ float sMax[32];
    __shared__ float vMax[32];

    const int c   = t & 31;
    const int grp = t >> 5;
    const float ws0 = W_s[c], ws1 = W_s[32 + c];
    const float wv0 = W_v[c], wv1 = W_v[32 + c];

    // Pass A: column maxima of p_s / p_v over valid j (NaN rows skipped; exact, see analysis)
    float sm = NEG_INF, vm = NEG_INF;
    for (int j = grp; j < N; j += 8) {
        const float x2 = obs2[(b * N + j) * 2 + 0];
        const float y2 = obs2[(b * N + j) * 2 + 1];
        const bool ok2 = !(x2 != x2) && !(y2 != y2);
        if (ok2) sm = fmaxf(sm, ws0 * x2 + ws1 * y2);
        const float x1 = obs1[(b * N + j) * 2 + 0];
        const float y1 = obs1[(b * N + j) * 2 + 1];
        const bool okv = ok2 && !(x1 != x1) && !(y1 != y1);
        if (okv) {
            const float vx = 4.0f * (x2 - x1);
            const float vy = 4.0f * (y2 - y1);
            vm = fmaxf(vm, wv0 * vx + wv1 * vy);
        }
    }
    sPart[grp][c] = sm;
    vPart[grp][c] = vm;
    __syncthreads();
    if (t < 32) {
        float s = sPart[0][t], v = vPart[0][t];
        for (int g = 1; g < 8; ++g) { s = fmaxf(s, sPart[g][t]); v = fmaxf(v, vPart[g][t]); }
        sMax[t] = s;
        vMax[t] = v;
    }
    __syncthreads();

    // Pass B: closed-form pooled values per query row i
    for (int i = t; i < N; i += 256) {
        const float x2 = obs2[(b * N + i) * 2 + 0];
        const float y2 = obs2[(b * N + i) * 2 + 1];
        const bool ok2 = !(x2 != x2) && !(y2 != y2);
        const float x1 = obs1[(b * N + i) * 2 + 0];
        const float y1 = obs1[(b * N + i) * 2 + 1];
        const bool okv = ok2 && !(x1 != x1) && !(y1 != y1);
        const float vx = 4.0f * (x2 - x1);
        const float vy = 4.0f * (y2 - y1);
        float* row = pooled + (size_t)(b * N + i) * 128;
        for (int cc = 0; cc < 32; ++cc) {
            const float ps = W_s[cc] * x2 + W_s[32 + cc] * y2;
            row[cc]      = ok2 ? fmaxf(sMax[cc] - ps + b_s[cc], 0.0f) : FILLV;
            const float pv = W_v[cc] * vx + W_v[32 + cc] * vy;
            row[96 + cc] = okv ? fmaxf(vMax[cc] - pv + b_v[cc], 0.0f) : FILLV;
        }
    }
}

// ---------------------------------------------------------------------------
// Kernel 2: hidden embed (WMMA f32 GEMM [512x128]x[128x64] per batch),
// bias+relu+NaN-row FILL, per-batch column max -> broadcast to cols [32,96).
// One block per batch, 8 waves; wave w owns jtiles 4w..4w+3.
// ---------------------------------------------------------------------------
__global__ void hidden_embed_max(const float* __restrict__ hs,
                                 const float* __restrict__ W_h,
                                 const float* __restrict__ b_h,
                                 float* __restrict__ pooled)
{
    const int b = blockIdx.x;
    const int t = threadIdx.x;
    const int N = 512, H = 128, DH = 64;

    __shared__ float rowBad[512];
    __shared__ float wmax[8][64];
    __shared__ float hmax[64];

    // Row NaN masks (any of 128 features NaN -> row becomes FILL after embed)
    for (int j = t; j < N; j += 256) {
        const float4* p = (const float4*)(hs + (size_t)(b * N + j) * H);
        bool bad = false;
        for (int q = 0; q < H / 4; ++q) {
            float4 f = p[q];
            bad = bad || (f.x != f.x) || (f.y != f.y) || (f.z != f.z) || (f.w != f.w);
        }
        rowBad[j] = bad ? 1.0f : 0.0f;
    }
    __syncthreads();

    const int wave = t >> 5;
    const int lane = t & 31;
    const int half = lane >> 4;      // 0: K lo pair / M 0-7, 1: K hi pair / M 8-15
    const int mrow = lane & 15;      // A row within tile
    const int ncol = lane & 15;      // B/C/D column within tile

    for (int nt = 0; nt < 4; ++nt) {                 // 4 n-tiles cover DH=64
        float cmax = NEG_INF;
        const float bias = b_h[nt * 16 + ncol];
        for (int jt4 = 0; jt4 < 4; ++jt4) {
            const int jt = wave * 4 + jt4;           // 32 jtiles cover N=512
            v8f acc = {};
            for (int k = 0; k < H; k += 4) {
                const int k2 = k + 2 * half;
                v2f a, bb;
                const float* ap = hs + (size_t)(b * N + jt * 16 + mrow) * H + k2;
                a.x  = ap[0];
                a.y  = ap[1];
                bb.x = W_h[(k2 + 0) * DH + nt * 16 + ncol];
                bb.y = W_h[(k2 + 1) * DH + nt * 16 + ncol];
                acc = __builtin_amdgcn_wmma_f32_16x16x4_f32(
                          false, a, false, bb, (short)0, acc, false, false);
            }
            for (int v = 0; v < 8; ++v) {
                const int row = jt * 16 + v + 8 * half;
                float val = fmaxf(acc[v] + bias, 0.0f);
                if (rowBad[row] != 0.0f) val = FILLV;
                cmax = fmaxf(cmax, val);
            }
        }
        cmax = fmaxf(cmax, __shfl_xor(cmax, 16, 32));   // merge M 0-7 / 8-15 halves
        if (half == 0) wmax[wave][nt * 16 + ncol] = cmax;
    }
    __syncthreads();
    if (t < 64) {
        float m = wmax[0][t];
        for (int w = 1; w < 8; ++w) m = fmaxf(m, wmax[w][t]);
        hmax[t] = m;
    }
    __syncthreads();
    // Hidden pooled part is identical for every query i -> broadcast
    for (int idx = t; idx < N * 64; idx += 256) {
        const int i = idx >> 6, cc = idx & 63;
        pooled[(size_t)(b * N + i) * 128 + 32 + cc] = hmax[cc];
    }
}

// ---------------------------------------------------------------------------
// Kernel 3: out = pooled[2048,128] @ W_o[128,128] + b_o, WMMA f32 16x16x4.
// 1024 16x16 tiles, one wave each; 8 waves/block, 128 blocks.
// ---------------------------------------------------------------------------
__global__ void out_gemm(const float* __restrict__ pooled,
                         const float* __restrict__ W_o,
                         const float* __restrict__ b_o,
                         float* __restrict__ out)
{
    const int wave = threadIdx.x >> 5;
    const int lane = threadIdx.x & 31;
    const int tile = blockIdx.x * 8 + wave;   // 0..1023
    const int mt = tile >> 3;                 // 128 m-tiles
    const int nt = tile & 7;                  // 8 n-tiles
    const int half = lane >> 4;
    const int mrow = lane & 15;
    const int ncol = lane & 15;
    const int K = 128, NO = 128;

    v8f acc = {};
    const float* arow = pooled + (size_t)(mt * 16 + mrow) * K;
    for (int k = 0; k < K; k += 4) {
        const int k2 = k + 2 * half;
        v2f a, bb;
        a.x  = arow[k2 + 0];
        a.y  = arow[k2 + 1];
        bb.x = W_o[(k2 + 0) * NO + nt * 16 + ncol];
        bb.y = W_o[(k2 + 1) * NO + nt * 16 + ncol];
        acc = __builtin_amdgcn_wmma_f32_16x16x4_f32(
                  false, a, false, bb, (short)0, acc, false, false);
    }
    const float bias = b_o[nt * 16 + ncol];
    for (int v = 0; v < 8; ++v) {
        const int r = mt * 16 + v + 8 * half;
        out[(size_t)r * NO + nt * 16 + ncol] = acc[v] + bias;
    }
}

extern "C" void kernel_launch(void* const* d_in, const int* in_sizes, int n_in,
                              void* d_out, int out_size, void* d_ws, size_t ws_size,
                              hipStream_t stream) {
    const float* hs   = (const float*)d_in[0];  // [4,512,128]
    const float* obs1 = (const float*)d_in[1];  // [4,512,2]
    const float* obs2 = (const float*)d_in[2];  // [4,512,2]
    const float* W_s  = (const float*)d_in[3];  // [2,32]
    const float* b_s  = (const float*)d_in[4];  // [32]
    const float* W_v  = (const float*)d_in[5];  // [2,32]
    const float* b_v  = (const float*)d_in[6];  // [32]
    const float* W_h  = (const float*)d_in[7];  // [128,64]
    const float* b_h  = (const float*)d_in[8];  // [64]
    const float* W_o  = (const float*)d_in[9];  // [128,128]
    const float* b_o  = (const float*)d_in[10]; // [128]
    float* out = (float*)d_out;                 // [2048,128]

    float* pooled = (float*)d_ws;               // 2048*128*4 = 1 MB scratch

    prep_pairwise   <<<4,   256, 0, stream>>>(obs1, obs2, W_s, b_s, W_v, b_v, pooled);
    hidden_embed_max<<<4,   256, 0, stream>>>(hs, W_h, b_h, pooled);
    out_gemm        <<<128, 256, 0, stream>>>(pooled, W_o, b_o, out);
}